// GPT2LinearSelfAttention_53283364274210
// MI455X (gfx1250) — compile-verified
//
#include <hip/hip_runtime.h>

// GPT-2 linear self-attention for MI455X (gfx1250, wave32, WMMA + async LDS).
// B=2, T=2048, C=512, H=8, D=64. fp32 throughout (V_WMMA_F32_16X16X4_F32).
//
// Pipeline:
//  1) qkv_phi_kernel : qkv = x@Wqkv+b, fused elu+1 / diag-mask, head scatter.
//                      Async-LDS double-buffered WMMA GEMM.
//  2a) chunk_kv_kernel : per (b,h,chunk) partial S_c = K_c^T @ V_c, kcsum.
//  2b) scan_kernel     : exclusive prefix over chunks (per b,h).
//  2c) chunk_out_kernel: O = tril(Q K^T) V + Q S_prev, denom, y = O/denom.
//  3) proj_kernel    : out = y@Wproj+b  (same GEMM skeleton as 1).

#define BATCH 2
#define TSEQ 2048
#define CDIM 512
#define NHEAD 8
#define DHEAD 64
#define QKVN 1536
#define NCHUNK 32              // TSEQ / 64
#define EPSV 1e-8f
#define KSTEP 32
#define APAD 36                // 32 + 4 (16B-aligned rows, conflict-free cols)
#define BPAD 68                // 64 + 4

typedef __attribute__((ext_vector_type(2))) float v2f;
typedef __attribute__((ext_vector_type(8))) float v8f;

__device__ __forceinline__ v8f zero8() {
  v8f z = {0.f, 0.f, 0.f, 0.f, 0.f, 0.f, 0.f, 0.f};
  return z;
}

// D = A(16x4 f32) * B(4x16 f32) + C(16x16 f32)
__device__ __forceinline__ v8f wmma4(v2f a, v2f b, v8f c) {
  return __builtin_amdgcn_wmma_f32_16x16x4_f32(
      false, a, false, b, (short)0, c, false, false);
}

// Async global->LDS copy, 16B per lane (VDST = LDS byte offset in a VGPR).
__device__ __forceinline__ void async_copy16(void* lds_ptr, const void* gptr) {
  unsigned loff = (unsigned)(unsigned long long)lds_ptr;  // low 32b = LDS offset
  asm volatile("global_load_async_to_lds_b128 %0, %1, off"
               :: "v"(loff), "v"((unsigned long long)gptr)
               : "memory");
}
__device__ __forceinline__ void wait_async0() {
  asm volatile("s_wait_asynccnt 0x0" ::: "memory");
}

__device__ __forceinline__ float phi_elu1(float x) {
  return x > 0.f ? x + 1.f : __expf(x);  // elu(x)+1
}

// ---------------------------------------------------------------------------
// Kernel 1: QKV GEMM (64x64 block tile, K-step 32, async-LDS double buffer).
// 8 waves: (wm in 0..1) x (wn in 0..3); wave -> rows [wm*32,+32), cols [wn*16,+16).
// grid = (QKVN/64 = 24, (B*T)/64 = 64)
// ---------------------------------------------------------------------------
__global__ __launch_bounds__(256) void qkv_phi_kernel(
    const float* __restrict__ x, const float* __restrict__ W,
    const float* __restrict__ bias, const float* __restrict__ amask,
    float* __restrict__ phiq, float* __restrict__ phik, float* __restrict__ vv)
{
  __shared__ float As[2][64][APAD];
  __shared__ float Bs[2][32][BPAD];

  const int tid = threadIdx.x;
  const int lane = tid & 31, wave = tid >> 5;
  const int wm = wave >> 2, wn = wave & 3;
  const int mr = lane & 15, hi = lane >> 4;
  const int m0 = blockIdx.y * 64, n0 = blockIdx.x * 64;

  auto stage = [&](int buf, int k0) {
#pragma unroll
    for (int i = 0; i < 2; ++i) {
      int idx4 = tid * 2 + i;                        // 0..511 16B slots
      int ar = idx4 >> 3, ac = (idx4 & 7) << 2;      // A: 64 x 8 float4
      async_copy16(&As[buf][ar][ac], x + (m0 + ar) * CDIM + k0 + ac);
      int br = idx4 >> 4, bc = (idx4 & 15) << 2;     // B: 32 x 16 float4
      async_copy16(&Bs[buf][br][bc], W + (k0 + br) * QKVN + n0 + bc);
    }
  };

  v8f acc0 = zero8(), acc1 = zero8();
  stage(0, 0);
  wait_async0();
  __syncthreads();

  int cur = 0;
  for (int t = 0; t < CDIM / KSTEP; ++t) {
    if (t + 1 < CDIM / KSTEP) stage(cur ^ 1, (t + 1) * KSTEP);
    const float(*Ac)[APAD] = As[cur];
    const float(*Bc)[BPAD] = Bs[cur];

    v2f a0c, a1c, bfc, a0n, a1n, bfn;
    {
      const int klo = hi << 1;
      a0c[0] = Ac[wm * 32 + mr][klo];      a0c[1] = Ac[wm * 32 + mr][klo + 1];
      a1c[0] = Ac[wm * 32 + 16 + mr][klo]; a1c[1] = Ac[wm * 32 + 16 + mr][klo + 1];
      bfc[0] = Bc[klo][wn * 16 + mr];      bfc[1] = Bc[klo + 1][wn * 16 + mr];
    }
#pragma unroll
    for (int kk = 0; kk < KSTEP; kk += 4) {
      if (kk + 4 < KSTEP) {  // prefetch next fragments while WMMA runs
        const int klo = kk + 4 + (hi << 1);
        a0n[0] = Ac[wm * 32 + mr][klo];      a0n[1] = Ac[wm * 32 + mr][klo + 1];
        a1n[0] = Ac[wm * 32 + 16 + mr][klo]; a1n[1] = Ac[wm * 32 + 16 + mr][klo + 1];
        bfn[0] = Bc[klo][wn * 16 + mr];      bfn[1] = Bc[klo + 1][wn * 16 + mr];
      }
      acc0 = wmma4(a0c, bfc, acc0);
      acc1 = wmma4(a1c, bfc, acc1);
      a0c = a0n; a1c = a1n; bfc = bfn;
    }
    wait_async0();
    __syncthreads();
    cur ^= 1;
  }

  // Epilogue: bias, phi, diagonal mask, scatter to [B,H,T,D]
#pragma unroll
  for (int t2 = 0; t2 < 2; ++t2) {
    v8f a = t2 ? acc1 : acc0;
#pragma unroll
    for (int r = 0; r < 8; ++r) {
      int ml = wm * 32 + t2 * 16 + r + (hi << 3);
      int nl = wn * 16 + mr;
      int m = m0 + ml, n = n0 + nl;
      float val = a[r] + bias[n];
      int b = m >> 11, t = m & (TSEQ - 1);
      int part = n >> 9, idx = n & 511;
      int h = idx >> 6, d = idx & 63;
      int off = ((b * NHEAD + h) * TSEQ + t) * DHEAD + d;
      if (part == 0) {
        phiq[off] = phi_elu1(val);
      } else {
        float mv = amask[b * TSEQ * TSEQ + t * TSEQ + t];
        if (part == 1) phik[off] = phi_elu1(val) * mv;
        else           vv[off]   = val * mv;
      }
    }
  }
}

// ---------------------------------------------------------------------------
// Kernel 2a: per-chunk S_c = K_c^T @ V_c (64x64) and kc = colsum(K_c).
// grid = (NCHUNK, B*H) = (32, 16)
// ---------------------------------------------------------------------------
__global__ __launch_bounds__(256) void chunk_kv_kernel(
    const float* __restrict__ phik, const float* __restrict__ vv,
    float* __restrict__ Scs,   // [B*H, NCHUNK, 64, 64]
    float* __restrict__ kcs)   // [B*H, NCHUNK, 64]
{
  __shared__ float Ks[64 * 64];  // row-major reads only -> no pad
  __shared__ float Vs[64 * 64];

  const int tid = threadIdx.x;
  const int lane = tid & 31, wave = tid >> 5;
  const int wm = wave >> 2, wn = wave & 3;
  const int mr = lane & 15, hi = lane >> 4;
  const int c = blockIdx.x, bh = blockIdx.y;
  const int base = bh * TSEQ * DHEAD + c * 64 * DHEAD;

#pragma unroll
  for (int i = 0; i < 4; ++i) {
    int off = (tid + i * 256) << 2;  // float offset, 16B chunks
    async_copy16(&Ks[off], phik + base + off);
    async_copy16(&Vs[off], vv + base + off);
  }
  wait_async0();
  __syncthreads();

  v8f s0 = zero8(), s1 = zero8();
#pragma unroll
  for (int kk = 0; kk < 64; kk += 4) {
    const int klo = kk + (hi << 1);
    v2f a0, a1, bf;  // A = K^T: A[d1][t] = K[t][d1] (row-major reads)
    a0[0] = Ks[klo * 64 + wm * 32 + mr];      a0[1] = Ks[(klo + 1) * 64 + wm * 32 + mr];
    a1[0] = Ks[klo * 64 + wm * 32 + 16 + mr]; a1[1] = Ks[(klo + 1) * 64 + wm * 32 + 16 + mr];
    bf[0] = Vs[klo * 64 + wn * 16 + mr];      bf[1] = Vs[(klo + 1) * 64 + wn * 16 + mr];
    s0 = wmma4(a0, bf, s0);
    s1 = wmma4(a1, bf, s1);
  }

  float* Sc = Scs + (bh * NCHUNK + c) * 4096;
#pragma unroll
  for (int r = 0; r < 8; ++r) {
    int d1 = wm * 32 + r + (hi << 3), d2 = wn * 16 + mr;
    Sc[d1 * 64 + d2]        = s0[r];
    Sc[(d1 + 16) * 64 + d2] = s1[r];
  }
  if (tid < 64) {
    float s = 0.f;
    for (int t = 0; t < 64; ++t) s += Ks[t * 64 + tid];
    kcs[(bh * NCHUNK + c) * 64 + tid] = s;
  }
}

// ---------------------------------------------------------------------------
// Kernel 2b: in-place exclusive prefix over chunks.  grid = B*H = 16
// ---------------------------------------------------------------------------
__global__ __launch_bounds__(256) void scan_kernel(float* __restrict__ Scs,
                                                   float* __restrict__ kcs)
{
  const int tid = threadIdx.x;
  const int bh = blockIdx.x;
  float acc[16];
#pragma unroll
  for (int j = 0; j < 16; ++j) acc[j] = 0.f;
  for (int c = 0; c < NCHUNK; ++c) {
    float* Sc = Scs + (bh * NCHUNK + c) * 4096;
#pragma unroll
    for (int j = 0; j < 16; ++j) {
      int idx = tid + j * 256;
      float t = Sc[idx];
      Sc[idx] = acc[j];
      acc[j] += t;
    }
  }
  if (tid < 64) {
    float a = 0.f;
    for (int c = 0; c < NCHUNK; ++c) {
      float* kc = kcs + (bh * NCHUNK + c) * 64;
      float t = kc[tid];
      kc[tid] = a;
      a += t;
    }
  }
}

// ---------------------------------------------------------------------------
// Kernel 2c: per-chunk output.
//  P = tril(Q K^T); O = P V + Q S_prev; denom_i = rowsum(P)_i + Q_i.kprev;
//  y = O / denom.   grid = (NCHUNK, B*H).  Dynamic LDS = 85504 B.
// ---------------------------------------------------------------------------
__global__ __launch_bounds__(256) void chunk_out_kernel(
    const float* __restrict__ phiq, const float* __restrict__ phik,
    const float* __restrict__ vv, const float* __restrict__ Scs,
    const float* __restrict__ kcs, float* __restrict__ y /* [B,T,C] */)
{
  extern __shared__ float sm[];
  float* Qs    = sm;               // 64 x BPAD (column reads across lanes)
  float* Ks    = Qs + 64 * BPAD;   // 64 x BPAD
  float* Ps    = Ks + 64 * BPAD;   // 64 x BPAD
  float* Vs    = Ps + 64 * BPAD;   // 64 x 64 (row reads only)
  float* Ss    = Vs + 64 * 64;     // 64 x 64 S_prev
  float* kprev = Ss + 64 * 64;     // 64
  float* denom = kprev + 64;       // 64

  const int tid = threadIdx.x;
  const int lane = tid & 31, wave = tid >> 5;
  const int wm = wave >> 2, wn = wave & 3;
  const int mr = lane & 15, hi = lane >> 4;
  const int c = blockIdx.x, bh = blockIdx.y;
  const int b = bh >> 3, h = bh & 7;
  const int t0 = c * 64;
  const int base = bh * TSEQ * DHEAD + t0 * DHEAD;
  const float* Sp = Scs + (bh * NCHUNK + c) * 4096;

#pragma unroll
  for (int i = 0; i < 4; ++i) {
    int idx4 = tid + i * 256;
    int r = idx4 >> 4, cc = (idx4 & 15) << 2;
    int g = base + r * DHEAD + cc;
    async_copy16(&Qs[r * BPAD + cc], phiq + g);
    async_copy16(&Ks[r * BPAD + cc], phik + g);
    async_copy16(&Vs[r * 64 + cc], vv + g);
    async_copy16(&Ss[idx4 << 2], Sp + (idx4 << 2));
  }
  if (tid < 16)
    async_copy16(&kprev[tid << 2], kcs + (bh * NCHUNK + c) * 64 + (tid << 2));
  wait_async0();
  __syncthreads();

  // P = tril(Q @ K^T)
  v8f p0 = zero8(), p1 = zero8();
#pragma unroll
  for (int kk = 0; kk < 64; kk += 4) {
    const int klo = kk + (hi << 1);
    v2f a0, a1, bf;
    a0[0] = Qs[(wm * 32 + mr) * BPAD + klo];      a0[1] = Qs[(wm * 32 + mr) * BPAD + klo + 1];
    a1[0] = Qs[(wm * 32 + 16 + mr) * BPAD + klo]; a1[1] = Qs[(wm * 32 + 16 + mr) * BPAD + klo + 1];
    bf[0] = Ks[(wn * 16 + mr) * BPAD + klo];      bf[1] = Ks[(wn * 16 + mr) * BPAD + klo + 1];
    p0 = wmma4(a0, bf, p0);
    p1 = wmma4(a1, bf, p1);
  }
#pragma unroll
  for (int r = 0; r < 8; ++r) {
    int i0 = wm * 32 + r + (hi << 3);
    int j = wn * 16 + mr;
    Ps[i0 * BPAD + j]        = (j <= i0)      ? p0[r] : 0.f;
    Ps[(i0 + 16) * BPAD + j] = (j <= i0 + 16) ? p1[r] : 0.f;
  }
  __syncthreads();

  // denom_i = rowsum(masked P)_i + Q_i . kprev   (phi >= 0)
  if (tid < 64) {
    float rs = 0.f, qd = 0.f;
    for (int j = 0; j < 64; ++j) rs += Ps[tid * BPAD + j];
    for (int d = 0; d < 64; ++d) qd += Qs[tid * BPAD + d] * kprev[d];
    float dn = rs + qd;
    denom[tid] = dn > EPSV ? dn : EPSV;
  }

  // O = P @ V + Q @ S_prev
  v8f o0 = zero8(), o1 = zero8();
#pragma unroll
  for (int kk = 0; kk < 64; kk += 4) {
    const int klo = kk + (hi << 1);
    v2f a0, a1, bf;
    a0[0] = Ps[(wm * 32 + mr) * BPAD + klo];      a0[1] = Ps[(wm * 32 + mr) * BPAD + klo + 1];
    a1[0] = Ps[(wm * 32 + 16 + mr) * BPAD + klo]; a1[1] = Ps[(wm * 32 + 16 + mr) * BPAD + klo + 1];
    bf[0] = Vs[klo * 64 + wn * 16 + mr];          bf[1] = Vs[(klo + 1) * 64 + wn * 16 + mr];
    o0 = wmma4(a0, bf, o0);
    o1 = wmma4(a1, bf, o1);
  }
#pragma unroll
  for (int kk = 0; kk < 64; kk += 4) {
    const int klo = kk + (hi << 1);
    v2f a0, a1, bf;
    a0[0] = Qs[(wm * 32 + mr) * BPAD + klo];      a0[1] = Qs[(wm * 32 + mr) * BPAD + klo + 1];
    a1[0] = Qs[(wm * 32 + 16 + mr) * BPAD + klo]; a1[1] = Qs[(wm * 32 + 16 + mr) * BPAD + klo + 1];
    bf[0] = Ss[klo * 64 + wn * 16 + mr];          bf[1] = Ss[(klo + 1) * 64 + wn * 16 + mr];
    o0 = wmma4(a0, bf, o0);
    o1 = wmma4(a1, bf, o1);
  }
  __syncthreads();  // denom visible

#pragma unroll
  for (int r = 0; r < 8; ++r) {
    int i0 = wm * 32 + r + (hi << 3);
    int n = wn * 16 + mr;
    int t = t0 + i0;
    y[(b * TSEQ + t) * CDIM + h * DHEAD + n]      = o0[r] / denom[i0];
    y[(b * TSEQ + t + 16) * CDIM + h * DHEAD + n] = o1[r] / denom[i0 + 16];
  }
}

// ---------------------------------------------------------------------------
// Kernel 3: out = y @ Wproj + bproj.  grid = (CDIM/64 = 8, (B*T)/64 = 64)
// ---------------------------------------------------------------------------
__global__ __launch_bounds__(256) void proj_kernel(
    const float* __restrict__ yin, const float* __restrict__ W,
    const float* __restrict__ bias, float* __restrict__ out)
{
  __shared__ float As[2][64][APAD];
  __shared__ float Bs[2][32][BPAD];

  const int tid = threadIdx.x;
  const int lane = tid & 31, wave = tid >> 5;
  const int wm = wave >> 2, wn = wave & 3;
  const int mr = lane & 15, hi = lane >> 4;
  const int m0 = blockIdx.y * 64, n0 = blockIdx.x * 64;

  auto stage = [&](int buf, int k0) {
#pragma unroll
    for (int i = 0; i < 2; ++i) {
      int idx4 = tid * 2 + i;
      int ar = idx4 >> 3, ac = (idx4 & 7) << 2;
      async_copy16(&As[buf][ar][ac], yin + (m0 + ar) * CDIM + k0 + ac);
      int br = idx4 >> 4, bc = (idx4 & 15) << 2;
      async_copy16(&Bs[buf][br][bc], W + (k0 + br) * CDIM + n0 + bc);
    }
  };

  v8f acc0 = zero8(), acc1 = zero8();
  stage(0, 0);
  wait_async0();
  __syncthreads();

  int cur = 0;
  for (int t = 0; t < CDIM / KSTEP; ++t) {
    if (t + 1 < CDIM / KSTEP) stage(cur ^ 1, (t + 1) * KSTEP);
    const float(*Ac)[APAD] = As[cur];
    const float(*Bc)[BPAD] = Bs[cur];

    v2f a0c, a1c, bfc, a0n, a1n, bfn;
    {
      const int klo = hi << 1;
      a0c[0] = Ac[wm * 32 + mr][klo];      a0c[1] = Ac[wm * 32 + mr][klo + 1];
      a1c[0] = Ac[wm * 32 + 16 + mr][klo]; a1c[1] = Ac[wm * 32 + 16 + mr][klo + 1];
      bfc[0] = Bc[klo][wn * 16 + mr];      bfc[1] = Bc[klo + 1][wn * 16 + mr];
    }
#pragma unroll
    for (int kk = 0; kk < KSTEP; kk += 4) {
      if (kk + 4 < KSTEP) {
        const int klo = kk + 4 + (hi << 1);
        a0n[0] = Ac[wm * 32 + mr][klo];      a0n[1] = Ac[wm * 32 + mr][klo + 1];
        a1n[0] = Ac[wm * 32 + 16 + mr][klo]; a1n[1] = Ac[wm * 32 + 16 + mr][klo + 1];
        bfn[0] = Bc[klo][wn * 16 + mr];      bfn[1] = Bc[klo + 1][wn * 16 + mr];
      }
      acc0 = wmma4(a0c, bfc, acc0);
      acc1 = wmma4(a1c, bfc, acc1);
      a0c = a0n; a1c = a1n; bfc = bfn;
    }
    wait_async0();
    __syncthreads();
    cur ^= 1;
  }

#pragma unroll
  for (int t2 = 0; t2 < 2; ++t2) {
    v8f a = t2 ? acc1 : acc0;
#pragma unroll
    for (int r = 0; r < 8; ++r) {
      int ml = wm * 32 + t2 * 16 + r + (hi << 3);
      int nl = wn * 16 + mr;
      int m = m0 + ml, n = n0 + nl;
      out[m * CDIM + n] = a[r] + bias[n];
    }
  }
}

// ---------------------------------------------------------------------------
extern "C" void kernel_launch(void* const* d_in, const int* in_sizes, int n_in,
                              void* d_out, int out_size, void* d_ws, size_t ws_size,
                              hipStream_t stream) {
  (void)in_sizes; (void)n_in; (void)out_size; (void)ws_size;

  const float* x     = (const float*)d_in[0];  // [B,T,C]
  const float* amask = (const float*)d_in[1];  // [B,T,T]
  const float* Wqkv  = (const float*)d_in[2];  // [C,3C]
  const float* bqkv  = (const float*)d_in[3];  // [3C]
  const float* Wproj = (const float*)d_in[4];  // [C,C]
  const float* bproj = (const float*)d_in[5];  // [C]
  float* out = (float*)d_out;

  const size_t HE = (size_t)BATCH * NHEAD * TSEQ * DHEAD;  // 2,097,152
  float* ws   = (float*)d_ws;
  float* phiq = ws;              // 8 MB
  float* phik = phiq + HE;       // 8 MB
  float* vv   = phik + HE;       // 8 MB
  float* yy   = vv + HE;         // 8 MB  [B,T,C]
  float* Scs  = yy + HE;         // 8 MB  [B*H, NCHUNK, 64, 64]
  float* kcs  = Scs + (size_t)BATCH * NHEAD * NCHUNK * 4096;  // 128 KB
  // total workspace use: ~40.1 MB

  qkv_phi_kernel<<<dim3(QKVN / 64, (BATCH * TSEQ) / 64), 256, 0, stream>>>(
      x, Wqkv, bqkv, amask, phiq, phik, vv);

  chunk_kv_kernel<<<dim3(NCHUNK, BATCH * NHEAD), 256, 0, stream>>>(
      phik, vv, Scs, kcs);

  scan_kernel<<<dim3(BATCH * NHEAD), 256, 0, stream>>>(Scs, kcs);

  const size_t lds2c = (size_t)(3 * 64 * BPAD + 2 * 64 * 64 + 128) * sizeof(float);
  chunk_out_kernel<<<dim3(NCHUNK, BATCH * NHEAD), 256, lds2c, stream>>>(
      phiq, phik, vv, Scs, kcs, yy);

  proj_kernel<<<dim3(CDIM / 64, (BATCH * TSEQ) / 64), 256, 0, stream>>>(
      yy, Wproj, bproj, out);
}